// AxialSpaceTimeTransformer_35098472742995
// MI455X (gfx1250) — compile-verified
//
#include <hip/hip_runtime.h>

typedef __attribute__((ext_vector_type(2))) float v2f;
typedef __attribute__((ext_vector_type(8))) float v8f;

#define D_    768
#define H_    12
#define DH_   64
#define DI_   768
#define DFF_  2048
#define B_    2
#define T_    32
#define S_    256
#define NTOK_ 16384   // B*T*S
#define CLAMP_ 50.0f
#define NEG_  -1.0e30f

__device__ __forceinline__ v8f wmma_f32(v2f a, v2f b, v8f c) {
  // D = A(16x4) * B(4x16) + C, fp32, wave32
  return __builtin_amdgcn_wmma_f32_16x16x4_f32(false, a, false, b, (short)0, c, false, false);
}

// 16-byte async copy: global -> LDS, tracked by ASYNCcnt (no VGPR staging).
__device__ __forceinline__ void async_copy_b128(unsigned int lds_byte_off,
                                                const float* gptr) {
  asm volatile("global_load_async_to_lds_b128 %0, %1, off"
               :: "v"(lds_byte_off), "v"((unsigned long long)(size_t)gptr)
               : "memory");
}

// ---------------------------------------------------------------- RMSNorm ---
__global__ __launch_bounds__(256) void rmsnorm_k(const float* __restrict__ x,
                                                 const float* __restrict__ g,
                                                 float* __restrict__ out) {
  __shared__ float red[8];
  const size_t row = blockIdx.x;
  const float* xr = x + row * D_;
  const int tid = threadIdx.x;
  float v0 = xr[tid], v1 = xr[tid + 256], v2 = xr[tid + 512];
  float s = v0 * v0 + v1 * v1 + v2 * v2;
#pragma unroll
  for (int off = 16; off; off >>= 1) s += __shfl_xor(s, off, 32);
  if ((tid & 31) == 0) red[tid >> 5] = s;
  __syncthreads();
  float tot = 0.f;
#pragma unroll
  for (int i = 0; i < 8; ++i) tot += red[i];
  const float r = rsqrtf(tot * (1.0f / (float)D_) + 1e-6f);
  out[row * D_ + tid]       = v0 * r * g[tid];
  out[row * D_ + tid + 256] = v1 * r * g[tid + 256];
  out[row * D_ + tid + 512] = v2 * r * g[tid + 512];
}

// ------------------------------------------------------------ WMMA f32 GEMM ---
// C[M,N] = A[M,K](lda) @ W[K,N] (+bias[N]) (+resid[M,N]).  M%64==0, N%32==0, K%32==0.
// Double-buffered LDS tiles filled by GLOBAL_LOAD_ASYNC_TO_LDS_B128 so the next
// tile's HBM fetch overlaps the current tile's WMMAs.
#define KC 32
#define ASTR 36   // padded LDS stride (floats): 16B-aligned rows, conflict-free cols
__global__ __launch_bounds__(256) void gemm_wmma(const float* __restrict__ A, int lda,
                                                 const float* __restrict__ W,
                                                 const float* __restrict__ bias,
                                                 const float* __restrict__ resid,
                                                 float* __restrict__ C,
                                                 int M, int K, int N) {
  __shared__ __align__(16) float As[2][64 * ASTR];
  __shared__ __align__(16) float Ws[2][32 * ASTR];
  const int tid  = threadIdx.x;
  const int lane = tid & 31;
  const int wave = tid >> 5;
  const int lrow = lane & 15;
  const int lhi  = lane >> 4;
  const int m0   = (wave & 3) * 16;
  const int n0   = (wave >> 2) * 16;
  const int bm   = blockIdx.y * 64;
  const int bn   = blockIdx.x * 32;

  // Per-thread staging coordinates (b128 = 4 floats per async op).
  // A tile 64x32 = 512 chunks -> 2 per thread; W tile 32x32 = 256 chunks -> 1.
  const int ar0 = tid >> 3;            // rows 0..31
  const int ar1 = ar0 + 32;            // rows 32..63
  const int ac  = (tid & 7) * 4;       // col within tile (16B aligned)

  const int NT = K / KC;
  v8f acc = {0.f, 0.f, 0.f, 0.f, 0.f, 0.f, 0.f, 0.f};

  auto issue = [&](int buf, int k0) {
    const unsigned int la0 = (unsigned int)(size_t)&As[buf][ar0 * ASTR + ac];
    const unsigned int la1 = (unsigned int)(size_t)&As[buf][ar1 * ASTR + ac];
    const unsigned int lw  = (unsigned int)(size_t)&Ws[buf][ar0 * ASTR + ac];
    async_copy_b128(la0, A + (size_t)(bm + ar0) * lda + (k0 + ac));
    async_copy_b128(la1, A + (size_t)(bm + ar1) * lda + (k0 + ac));
    async_copy_b128(lw,  W + (size_t)(k0 + ar0) * N + (bn + ac));
  };

  issue(0, 0);
  for (int it = 0; it < NT; ++it) {
    const int cur = it & 1;
    const bool more = (it + 1) < NT;
    if (more) issue(cur ^ 1, (it + 1) * KC);   // prefetch next tile into free buffer
    if (more) asm volatile("s_wait_asynccnt 3" ::: "memory");  // tile `it` resident
    else      asm volatile("s_wait_asynccnt 0" ::: "memory");
    __syncthreads();
#pragma unroll
    for (int kk = 0; kk < KC; kk += 4) {
      const int kb = kk + lhi * 2;
      v2f a, b;
      a.x = As[cur][(m0 + lrow) * ASTR + kb];
      a.y = As[cur][(m0 + lrow) * ASTR + kb + 1];
      b.x = Ws[cur][kb * ASTR + n0 + lrow];
      b.y = Ws[cur][(kb + 1) * ASTR + n0 + lrow];
      acc = wmma_f32(a, b, acc);
    }
    __syncthreads();   // all waves done reading `cur` before it is re-filled
  }
  const int gn = bn + n0 + lrow;
  const float badd = bias ? bias[gn] : 0.f;
#pragma unroll
  for (int r = 0; r < 8; ++r) {
    const int gm = bm + m0 + r + 8 * lhi;
    float v = acc[r] + badd;
    if (resid) v += resid[(size_t)gm * N + gn];
    C[(size_t)gm * N + gn] = v;
  }
}

// ------------------------------------------------------------ WMMA attention ---
// One wave per (pair, head, 16-query tile).  mode 0: space (L=256, special-token
// mask, stride 1);  mode 1: time (L=32, causal, stride S, rope pre-applied).
__global__ __launch_bounds__(32) void attn_wmma(const float* __restrict__ qkv,
                                                float* __restrict__ o, int mode) {
  __shared__ float Qs[16 * 65];
  __shared__ float Ks[16 * 65];
  __shared__ float Ps[16 * 257];
  const int lane = threadIdx.x;
  const int lrow = lane & 15;
  const int lhi  = lane >> 4;
  const int p = blockIdx.x, h = blockIdx.y;
  const int qi0 = blockIdx.z * 16;
  size_t tok0; int stride, L;
  if (mode == 0) { tok0 = (size_t)p * S_; stride = 1;  L = S_; }
  else { const int b = p >> 8, s = p & 255; tok0 = (size_t)b * (T_ * S_) + s; stride = S_; L = T_; }
  const int qoff = h * DH_;
  for (int i = lane; i < 16 * DH_; i += 32) {
    const int r = i >> 6, d = i & 63;
    Qs[r * 65 + d] = qkv[(tok0 + (size_t)(qi0 + r) * stride) * (3 * DI_) + qoff + d];
  }
  __syncthreads();
  const float scale = 0.125f;  // DH^-0.5
  for (int j0 = 0; j0 < L; j0 += 16) {
    for (int i = lane; i < 16 * DH_; i += 32) {
      const int r = i >> 6, d = i & 63;
      Ks[r * 65 + d] = qkv[(tok0 + (size_t)(j0 + r) * stride) * (3 * DI_) + DI_ + qoff + d];
    }
    __syncthreads();
    v8f acc = {0.f, 0.f, 0.f, 0.f, 0.f, 0.f, 0.f, 0.f};
#pragma unroll
    for (int kk = 0; kk < DH_; kk += 4) {
      const int kb = kk + lhi * 2;
      v2f a, b;
      a.x = Qs[lrow * 65 + kb]; a.y = Qs[lrow * 65 + kb + 1];
      b.x = Ks[lrow * 65 + kb]; b.y = Ks[lrow * 65 + kb + 1];
      acc = wmma_f32(a, b, acc);
    }
#pragma unroll
    for (int r = 0; r < 8; ++r) {
      const int mi = r + 8 * lhi;
      const int i  = qi0 + mi;
      const int j  = j0 + lrow;
      float sc = acc[r] * scale;
      sc = CLAMP_ * tanhf(sc * (1.0f / CLAMP_));
      const bool ok = mode ? (j <= i) : ((j < S_ - 1) || (i == j));
      Ps[mi * 257 + j] = ok ? sc : NEG_;
    }
    __syncthreads();
  }
  // softmax over row lrow: lane pair (lrow, lrow+16) split even/odd columns
  float mx = -3.0e38f;
  for (int j = lhi; j < L; j += 2) mx = fmaxf(mx, Ps[lrow * 257 + j]);
  mx = fmaxf(mx, __shfl_xor(mx, 16, 32));
  float sum = 0.f;
  for (int j = lhi; j < L; j += 2) {
    const float e = __expf(Ps[lrow * 257 + j] - mx);
    Ps[lrow * 257 + j] = e;
    sum += e;
  }
  sum += __shfl_xor(sum, 16, 32);
  const float inv = 1.0f / sum;
  for (int j = lhi; j < L; j += 2) Ps[lrow * 257 + j] *= inv;
  __syncthreads();
  // O(16x64) = P(16xL) @ V(Lx64)
  const v8f z8 = {0.f, 0.f, 0.f, 0.f, 0.f, 0.f, 0.f, 0.f};
  v8f oacc[4] = {z8, z8, z8, z8};
  for (int j0 = 0; j0 < L; j0 += 16) {
    for (int i = lane; i < 16 * DH_; i += 32) {
      const int r = i >> 6, d = i & 63;
      Ks[r * 65 + d] = qkv[(tok0 + (size_t)(j0 + r) * stride) * (3 * DI_) + 2 * DI_ + qoff + d];
    }
    __syncthreads();
#pragma unroll
    for (int kk = 0; kk < 16; kk += 4) {
      const int kb = kk + lhi * 2;
      v2f a;
      a.x = Ps[lrow * 257 + j0 + kb];
      a.y = Ps[lrow * 257 + j0 + kb + 1];
#pragma unroll
      for (int nt = 0; nt < 4; ++nt) {
        v2f b;
        b.x = Ks[kb * 65 + nt * 16 + lrow];
        b.y = Ks[(kb + 1) * 65 + nt * 16 + lrow];
        oacc[nt] = wmma_f32(a, b, oacc[nt]);
      }
    }
    __syncthreads();
  }
#pragma unroll
  for (int nt = 0; nt < 4; ++nt)
#pragma unroll
    for (int r = 0; r < 8; ++r) {
      const int mi = r + 8 * lhi;
      o[(tok0 + (size_t)(qi0 + mi) * stride) * DI_ + qoff + nt * 16 + lrow] = oacc[nt][r];
    }
}

// ------------------------------------------------------- mix: v = lerp(rv,v,σ) ---
__global__ __launch_bounds__(256) void mix_k(const float* __restrict__ n,
                                             const float* __restrict__ wmix,
                                             const float* __restrict__ bmix,
                                             const float* __restrict__ rv,
                                             float* __restrict__ qkv) {
  const int wid  = blockIdx.x * 8 + (threadIdx.x >> 5);  // (tok, head)
  const int lane = threadIdx.x & 31;
  const size_t tok = (size_t)wid / H_;
  const int h = wid - (int)(tok * H_);
  float s = 0.f;
  for (int k = lane; k < D_; k += 32) s += n[tok * D_ + k] * wmix[(size_t)k * H_ + h];
#pragma unroll
  for (int off = 16; off; off >>= 1) s += __shfl_xor(s, off, 32);
  const float m = 1.0f / (1.0f + __expf(-(s + bmix[h])));
  const size_t vb = tok * (3 * DI_) + 2 * DI_ + (size_t)h * DH_;
  const size_t rb = tok * DI_ + (size_t)h * DH_;
  for (int d = lane; d < DH_; d += 32) {
    const float v = qkv[vb + d];
    qkv[vb + d] = v * m + rv[rb + d] * (1.0f - m);
  }
}

// ------------------------------------------------------------------- RoPE ---
__global__ __launch_bounds__(256) void rope_k(float* __restrict__ qkv) {
  const int idx = blockIdx.x * 256 + threadIdx.x;  // tok*H*32 + h*32 + j
  const int j = idx & 31;
  const int h = (idx >> 5) % H_;
  const size_t tok = (size_t)idx / (32 * H_);
  const int t = (int)((tok / S_) % T_);
  const float invf = __powf(10000.0f, -(float)(2 * j) * (1.0f / (float)DH_));
  const float ang = (float)t * invf;
  float sn, c;
  __sincosf(ang, &sn, &c);
  size_t base = tok * (3 * DI_) + (size_t)h * DH_;
#pragma unroll
  for (int part = 0; part < 2; ++part) {  // q then k
    const float x1 = qkv[base + j];
    const float x2 = qkv[base + 32 + j];
    qkv[base + j]      = x1 * c - x2 * sn;
    qkv[base + 32 + j] = x2 * c + x1 * sn;
    base += DI_;
  }
}

// -------------------------------------------------------------- SwiGLU gate ---
__global__ __launch_bounds__(256) void silu_k(float* __restrict__ hbuf) {
  const size_t idx = (size_t)blockIdx.x * 256 + threadIdx.x;  // row*DFF + col
  const size_t row = idx / DFF_;
  const int col = (int)(idx - row * DFF_);
  float* hr = hbuf + row * (2 * DFF_);
  const float a = hr[col];
  const float g = hr[DFF_ + col];
  hr[col] = a * (g / (1.0f + __expf(-g)));
}

// ---------------------------------------------------------------- GRU gate ---
__global__ __launch_bounds__(256) void gru_gate_k(const float* __restrict__ xi,
                                                  const float* __restrict__ hh,
                                                  float* __restrict__ h,
                                                  float* __restrict__ x, int t) {
  const int idx = blockIdx.x * 256 + threadIdx.x;  // p*D + d, p = b*S + s
  const int p = idx / D_;
  const int d = idx - p * D_;
  const int b = p / S_;
  const int s = p - b * S_;
  const size_t tok = (size_t)(b * T_ + t) * S_ + s;
  const float xr = xi[tok * (3 * D_) + d];
  const float xz = xi[tok * (3 * D_) + D_ + d];
  const float xn = xi[tok * (3 * D_) + 2 * D_ + d];
  const float hr = hh[(size_t)p * (3 * D_) + d];
  const float hz = hh[(size_t)p * (3 * D_) + D_ + d];
  const float hn = hh[(size_t)p * (3 * D_) + 2 * D_ + d];
  const float r = 1.0f / (1.0f + __expf(-(xr + hr)));
  const float z = 1.0f / (1.0f + __expf(-(xz + hz)));
  const float cand = tanhf(xn + r * hn);
  const float hnew = (1.0f - z) * cand + z * h[idx];
  h[idx] = hnew;
  x[tok * D_ + d] += hnew;
}

// ------------------------------------------------------------------- host ---
extern "C" void kernel_launch(void* const* d_in, const int* in_sizes, int n_in,
                              void* d_out, int out_size, void* d_ws, size_t ws_size,
                              hipStream_t stream) {
  (void)in_sizes; (void)n_in; (void)out_size; (void)ws_size;
  const float* tokens = (const float*)d_in[0];
  struct LayerP {
    const float *anorm, *wqkv, *wmix, *bmix, *wout;
    const float *fnorm, *win, *fwout;
    const float *rnorm, *wih, *whh, *bih, *bhh;
    bool rnn;
  } L[8];
  int idx = 1;
  for (int i = 0; i < 8; ++i) {
    L[i].anorm = (const float*)d_in[idx++];
    L[i].wqkv  = (const float*)d_in[idx++];
    L[i].wmix  = (const float*)d_in[idx++];
    L[i].bmix  = (const float*)d_in[idx++];
    L[i].wout  = (const float*)d_in[idx++];
    L[i].fnorm = (const float*)d_in[idx++];
    L[i].win   = (const float*)d_in[idx++];
    L[i].fwout = (const float*)d_in[idx++];
    L[i].rnn = ((i + 1) % 4 == 0);
    L[i].rnorm = L[i].wih = L[i].whh = L[i].bih = L[i].bhh = nullptr;
    if (L[i].rnn) {
      L[i].rnorm = (const float*)d_in[idx++];
      L[i].wih   = (const float*)d_in[idx++];
      L[i].whh   = (const float*)d_in[idx++];
      L[i].bih   = (const float*)d_in[idx++];
      L[i].bhh   = (const float*)d_in[idx++];
    }
  }
  const float* vr_norm  = (const float*)d_in[idx++];
  const float* vr_w     = (const float*)d_in[idx++];
  const float* fin_norm = (const float*)d_in[idx++];

  float* x    = (float*)d_out;                      // activations (B,T,S,D)
  float* nbuf = (float*)d_ws;                       // NTOK*D   (norm out / attn out)
  float* qkvb = nbuf + (size_t)NTOK_ * D_;          // NTOK*3DI (also GRU xi)
  float* rvb  = qkvb + (size_t)NTOK_ * (3 * DI_);   // NTOK*DI  (residual value stream)
  float* hbuf = rvb  + (size_t)NTOK_ * DI_;         // NTOK*2DFF (FF hidden)
  float* hst  = hbuf + (size_t)NTOK_ * (2 * DFF_);  // B*S*D     (GRU state)
  float* hhb  = hst  + (size_t)(B_ * S_) * D_;      // B*S*3D    (GRU h@Whh)

  auto gemm = [&](const float* A, int lda, const float* W, int N, const float* bias,
                  const float* resid, float* C, int M, int K) {
    dim3 g(N / 32, M / 64);
    gemm_wmma<<<g, 256, 0, stream>>>(A, lda, W, bias, resid, C, M, K, N);
  };

  // rv = rmsnorm(tokens, vr_norm) @ vr_w ; x = tokens
  rmsnorm_k<<<NTOK_, 256, 0, stream>>>(tokens, vr_norm, nbuf);
  gemm(nbuf, D_, vr_w, DI_, nullptr, nullptr, rvb, NTOK_, D_);
  hipMemcpyAsync(x, tokens, (size_t)NTOK_ * D_ * sizeof(float),
                 hipMemcpyDeviceToDevice, stream);

  for (int i = 0; i < 8; ++i) {
    if (L[i].rnn) {  // GRU over T (time-major view is pointwise except the scan)
      rmsnorm_k<<<NTOK_, 256, 0, stream>>>(x, L[i].rnorm, nbuf);
      gemm(nbuf, D_, L[i].wih, 3 * D_, L[i].bih, nullptr, qkvb, NTOK_, D_);
      hipMemsetAsync(hst, 0, (size_t)(B_ * S_) * D_ * sizeof(float), stream);
      for (int t = 0; t < T_; ++t) {
        gemm(hst, D_, L[i].whh, 3 * D_, L[i].bhh, nullptr, hhb, B_ * S_, D_);
        gru_gate_k<<<(B_ * S_ * D_) / 256, 256, 0, stream>>>(qkvb, hhb, hst, x, t);
      }
    }
    // attention block
    rmsnorm_k<<<NTOK_, 256, 0, stream>>>(x, L[i].anorm, nbuf);
    gemm(nbuf, D_, L[i].wqkv, 3 * DI_, nullptr, nullptr, qkvb, NTOK_, D_);
    if (L[i].rnn)
      rope_k<<<(NTOK_ * H_ * 32) / 256, 256, 0, stream>>>(qkvb);
    mix_k<<<(NTOK_ * H_) / 8, 256, 0, stream>>>(nbuf, L[i].wmix, L[i].bmix, rvb, qkvb);
    if (L[i].rnn)
      attn_wmma<<<dim3(B_ * S_, H_, T_ / 16), 32, 0, stream>>>(qkvb, nbuf, 1);
    else
      attn_wmma<<<dim3(B_ * T_, H_, S_ / 16), 32, 0, stream>>>(qkvb, nbuf, 0);
    gemm(nbuf, DI_, L[i].wout, D_, nullptr, x, x, NTOK_, DI_);
    // feed-forward block
    rmsnorm_k<<<NTOK_, 256, 0, stream>>>(x, L[i].fnorm, nbuf);
    gemm(nbuf, D_, L[i].win, 2 * DFF_, nullptr, nullptr, hbuf, NTOK_, D_);
    silu_k<<<((size_t)NTOK_ * DFF_) / 256, 256, 0, stream>>>(hbuf);
    gemm(hbuf, 2 * DFF_, L[i].fwout, D_, nullptr, x, x, NTOK_, DFF_);
  }
  rmsnorm_k<<<NTOK_, 256, 0, stream>>>(x, fin_norm, x);
}